// GCN_layer_37520834297961
// MI455X (gfx1250) — compile-verified
//
#include <hip/hip_runtime.h>

typedef __attribute__((ext_vector_type(2))) float v2f;
typedef __attribute__((ext_vector_type(8))) float v8f;

#define IN_F 128
#define OUT_F 128
#define LDS_PITCH 130   // 128 + 2 pad: bank = (2*n) mod 64 -> conflict-free across 16 lanes

// -----------------------------------------------------------------------------
// Kernel 1: x = A @ W^T + b using V_WMMA_F32_16X16X4_F32 (exact fp32 path).
// Block = 128 threads = 4 waves. Each wave computes one 16-row tile across all
// 128 output columns (8 accumulator tiles). W staged in padded LDS once/block.
// -----------------------------------------------------------------------------
__global__ __launch_bounds__(128) void gemm_wmma_f32(
    const float* __restrict__ A,      // [N, 128]
    const float* __restrict__ W,      // [128, 128] row-major (out_f, in_f)
    const float* __restrict__ bias,   // [128]
    float* __restrict__ X,            // [N, 128]
    int nRowTiles)
{
    __shared__ float Wl[OUT_F * LDS_PITCH];

    // Cooperative stage of W into LDS (padded pitch).
    for (int i = threadIdx.x; i < OUT_F * IN_F; i += blockDim.x) {
        int n = i >> 7;           // /128
        int k = i & 127;
        Wl[n * LDS_PITCH + k] = W[i];
    }
    __syncthreads();

    const int wave = threadIdx.x >> 5;
    const int lane = threadIdx.x & 31;
    const int tile = blockIdx.x * 4 + wave;
    if (tile >= nRowTiles) return;

    const int row0 = tile * 16;
    const int m    = lane & 15;          // A row within tile / B col within tile
    const int kh   = (lane >> 4) << 1;   // lane-half picks K pair {0,1} vs {2,3}

    v8f acc[8] = {};

    const float* arow = A + (size_t)(row0 + m) * IN_F;

    #pragma unroll 4
    for (int kk = 0; kk < IN_F; kk += 4) {
        // A fragment: 16x4 fp32, 2 VGPRs/lane (ISA 7.12.2 layout) -> one b64 load
        v2f afrag = *(const v2f*)(arow + kk + kh);
        #pragma unroll
        for (int t = 0; t < 8; ++t) {
            // B fragment: B[k][n] = W[n][k]; same lane-half K-pair layout
            v2f bfrag = *(const v2f*)(&Wl[(t * 16 + m) * LDS_PITCH + kk + kh]);
            acc[t] = __builtin_amdgcn_wmma_f32_16x16x4_f32(
                false, afrag, false, bfrag, (short)0, acc[t], false, false);
        }
    }

    // C/D layout: VGPR r -> rows {r, r+8}; lanes 0-15 -> N=lane, 16-31 -> N=lane-16
    const int nOff = lane & 15;
    const int mOff = (lane >> 4) * 8;
    #pragma unroll
    for (int t = 0; t < 8; ++t) {
        const int   col = t * 16 + nOff;
        const float bv  = bias[col];
        #pragma unroll
        for (int r = 0; r < 8; ++r) {
            X[(size_t)(row0 + mOff + r) * OUT_F + col] = acc[t][r] + bv;
        }
    }
}

// -----------------------------------------------------------------------------
// Kernel 2: COO SpMM scatter: out[rows[e]] += vals[e] * x[cols[e]].
// One wave32 per edge; lane owns a float4 slice (32*4 = 128 = OUT_F).
// x and out both live in the 192 MB L2, so gathers + atomics stay on-die.
// -----------------------------------------------------------------------------
__global__ __launch_bounds__(256) void spmm_atomic(
    const float* __restrict__ X,      // [N, 128]
    const int*   __restrict__ rows,   // [E]
    const int*   __restrict__ cols,   // [E]
    const float* __restrict__ vals,   // [E]
    float* __restrict__ out,          // [N, 128]
    int E)
{
    const int lane = threadIdx.x & 31;
    const int wid  = (blockIdx.x * blockDim.x + threadIdx.x) >> 5;
    const int nw   = (gridDim.x * blockDim.x) >> 5;

    for (int e = wid; e < E; e += nw) {
        if (e + nw < E) {  // prefetch next iteration's metadata (global_prefetch_b8)
            __builtin_prefetch(&rows[e + nw], 0, 0);
            __builtin_prefetch(&cols[e + nw], 0, 0);
            __builtin_prefetch(&vals[e + nw], 0, 0);
        }
        const int   r = rows[e];
        const int   c = cols[e];
        const float v = vals[e];

        const float4 xv = ((const float4*)(X + (size_t)c * OUT_F))[lane];
        float* o = out + (size_t)r * OUT_F + lane * 4;
        atomicAdd(o + 0, v * xv.x);
        atomicAdd(o + 1, v * xv.y);
        atomicAdd(o + 2, v * xv.z);
        atomicAdd(o + 3, v * xv.w);
    }
}

// -----------------------------------------------------------------------------
// Launch
// -----------------------------------------------------------------------------
extern "C" void kernel_launch(void* const* d_in, const int* in_sizes, int n_in,
                              void* d_out, int out_size, void* d_ws, size_t ws_size,
                              hipStream_t stream)
{
    const float* layer_input = (const float*)d_in[0];
    const int*   adj_rows    = (const int*)  d_in[1];
    const int*   adj_cols    = (const int*)  d_in[2];
    const float* adj_vals    = (const float*)d_in[3];
    const float* W           = (const float*)d_in[4];
    const float* bias        = (const float*)d_in[5];

    const int N = in_sizes[0] / IN_F;     // 100000
    const int E = in_sizes[1];            // 6400000

    float* X   = (float*)d_ws;            // [N, OUT_F] intermediate (51.2 MB)
    float* out = (float*)d_out;

    // 1) GEMM: x = A @ W^T + b   (N is a multiple of 16: 100000/16 = 6250)
    const int nRowTiles = (N + 15) / 16;
    const int gemmBlocks = (nRowTiles + 3) / 4;   // 4 waves per block
    hipLaunchKernelGGL(gemm_wmma_f32, dim3(gemmBlocks), dim3(128), 0, stream,
                       layer_input, W, bias, X, nRowTiles);

    // 2) Zero the output accumulator (graph-capture-safe async memset)
    hipMemsetAsync(d_out, 0, (size_t)N * OUT_F * sizeof(float), stream);

    // 3) SpMM scatter-add
    const int spmmBlocks = 4096;                  // 32768 waves, ~195 edges each
    hipLaunchKernelGGL(spmm_atomic, dim3(spmmBlocks), dim3(256), 0, stream,
                       X, adj_rows, adj_cols, adj_vals, out, E);
}